// MSDeformAttn_46420006535953
// MI455X (gfx1250) — compile-verified
//
#include <hip/hip_runtime.h>

// MS-Deformable-Attention forward for MI455X (gfx1250, wave32).
// Gather-bound op: value[] (41MB) fits in 192MB L2; lane==channel gives
// 128B-coalesced gathers. CDNA5-specific path used: async global->LDS
// staging of sampling locations / attention weights with double buffering
// and s_wait_asynccnt (TDM/WMMA do not apply: no shared matmul operand).

#define HD   8
#define CH   32
#define NLVL 4
#define NP   4
#define LOC_PER_BQ (HD * NLVL * NP * 2)  // 256 floats per (b,q)
#define AW_PER_BQ  (HD * NLVL * NP)      // 128 floats per (b,q)

#if defined(__has_builtin)
#if __has_builtin(__builtin_amdgcn_global_load_async_to_lds_b32) && \
    __has_builtin(__builtin_amdgcn_s_wait_asynccnt)
#define USE_ASYNC 1
#endif
#endif
#ifndef USE_ASYNC
#define USE_ASYNC 0
#endif

typedef __attribute__((address_space(1))) int* gptr_i32;
typedef __attribute__((address_space(3))) int* lptr_i32;

// Stage this wave's loc (32 floats) + attention weights (16 floats) for
// query-block `bq` into LDS. Entirely wave-local: the data a wave reads back
// is the data the same wave loaded, so no workgroup barrier is required —
// only a per-wave s_wait_asynccnt.
__device__ __forceinline__ void stage_bq(const float* __restrict__ loc,
                                         const float* __restrict__ aw,
                                         float* slc, float* saw,
                                         unsigned bq, int h, int lane)
{
  const float* gl = loc + (size_t)bq * LOC_PER_BQ + h * 32 + lane;
  const float* ga = aw  + (size_t)bq * AW_PER_BQ  + h * 16 + (lane & 15);
  float* ll = slc + h * 32 + lane;
  float* la = saw + h * 16 + (lane & 15);
#if USE_ASYNC
  __builtin_amdgcn_global_load_async_to_lds_b32(
      (gptr_i32)gl, (lptr_i32)ll, 0, 0);
  if (lane < 16) {
    __builtin_amdgcn_global_load_async_to_lds_b32(
        (gptr_i32)ga, (lptr_i32)la, 0, 0);
  }
#else
  *ll = *gl;
  if (lane < 16) *la = *ga;
#endif
}

__global__ __launch_bounds__(256)
void MSDeformAttn_gfx1250_kernel(const float* __restrict__ value,
                                 const int*   __restrict__ shapes,
                                 const int*   __restrict__ starts,
                                 const float* __restrict__ loc,
                                 const float* __restrict__ aw,
                                 float* __restrict__ out,
                                 unsigned Q, unsigned L, unsigned nbq)
{
  __shared__ float s_loc[2][LOC_PER_BQ];
  __shared__ float s_aw [2][AW_PER_BQ];

  const int tid  = (int)threadIdx.x;
  const int lane = tid & 31;   // channel c
  const int h    = tid >> 5;   // head 0..7 (one wave32 per head)

  // Level geometry: uniform addresses -> scalar loads through K$.
  int Hdim[NLVL], Wdim[NLVL], Sdim[NLVL];
#pragma unroll
  for (int l = 0; l < NLVL; ++l) {
    Hdim[l] = shapes[2 * l];
    Wdim[l] = shapes[2 * l + 1];
    Sdim[l] = starts[l];
  }

  unsigned bq = blockIdx.x;
  if (bq >= nbq) return;

  stage_bq(loc, aw, s_loc[0], s_aw[0], bq, h, lane);

  int buf = 0;
  const unsigned stride = gridDim.x;
  for (; bq < nbq; bq += stride, buf ^= 1) {
    // Prefetch next iteration's loc/weights into the other LDS buffer.
    unsigned nxt = bq + stride;
    if (nxt >= nbq) nxt = bq;  // harmless duplicate fetch on the tail
    stage_bq(loc, aw, s_loc[buf ^ 1], s_aw[buf ^ 1], nxt, h, lane);

#if USE_ASYNC
    // 2 async instrs just issued may stay outstanding; wait for the 2
    // belonging to the current buffer to complete.
    __builtin_amdgcn_s_wait_asynccnt(2);
    asm volatile("" ::: "memory");
#endif

    const float locv = s_loc[buf][h * 32 + lane];        // lane i: loc elem i
    const float awv  = s_aw [buf][h * 16 + (lane & 15)]; // lane j<16: aw elem j

    const unsigned b = bq / Q;  // B==2, cheap 32-bit udiv once per 16 samples
    float acc = 0.0f;

#pragma unroll
    for (int j = 0; j < NLVL * NP; ++j) {
      const int l = j >> 2;
      const int W = Wdim[l];
      const int H = Hdim[l];

      const float x   = __shfl(locv, 2 * j)     * (float)W - 0.5f;
      const float y   = __shfl(locv, 2 * j + 1) * (float)H - 0.5f;
      const float wgt = __shfl(awv,  j);

      const float xf = floorf(x);
      const float yf = floorf(y);
      const float dx = x - xf;
      const float dy = y - yf;
      const int ix = (int)xf;
      const int iy = (int)yf;

      // zero-padded border handling, matching the reference
      const float vx0 = (ix     >= 0 && ix     < W) ? 1.0f : 0.0f;
      const float vx1 = (ix + 1 >= 0 && ix + 1 < W) ? 1.0f : 0.0f;
      const float vy0 = (iy     >= 0 && iy     < H) ? 1.0f : 0.0f;
      const float vy1 = (iy + 1 >= 0 && iy + 1 < H) ? 1.0f : 0.0f;

      const float w00 = (1.0f - dy) * (1.0f - dx) * vy0 * vx0;
      const float w01 = (1.0f - dy) * dx          * vy0 * vx1;
      const float w10 = dy          * (1.0f - dx) * vy1 * vx0;
      const float w11 = dy          * dx          * vy1 * vx1;

      const int cx0 = min(max(ix,     0), W - 1);
      const int cx1 = min(max(ix + 1, 0), W - 1);
      const int cy0 = min(max(iy,     0), H - 1);
      const int cy1 = min(max(iy + 1, 0), H - 1);

      // value[b, s+idx, h, c]; lane==c -> 128B coalesced global_load_b32
      const float* vb =
          value + (((size_t)b * L + (unsigned)Sdim[l]) * HD + h) * CH + lane;
      const float g00 = vb[(size_t)(cy0 * W + cx0) * (HD * CH)];
      const float g01 = vb[(size_t)(cy0 * W + cx1) * (HD * CH)];
      const float g10 = vb[(size_t)(cy1 * W + cx0) * (HD * CH)];
      const float g11 = vb[(size_t)(cy1 * W + cx1) * (HD * CH)];

      const float s =
          fmaf(w00, g00, fmaf(w01, g01, fmaf(w10, g10, w11 * g11)));
      acc = fmaf(wgt, s, acc);
    }

    // out[b, q, h*32 + c] : 128B coalesced store per wave
    out[((size_t)bq * HD + h) * CH + lane] = acc;
  }

#if USE_ASYNC
  __builtin_amdgcn_s_wait_asynccnt(0);
#endif
}

extern "C" void kernel_launch(void* const* d_in, const int* in_sizes, int n_in,
                              void* d_out, int out_size, void* d_ws, size_t ws_size,
                              hipStream_t stream) {
  (void)n_in; (void)out_size; (void)d_ws; (void)ws_size;

  const float* value  = (const float*)d_in[0];
  const int*   shapes = (const int*)d_in[1];
  const int*   starts = (const int*)d_in[2];
  const float* loc    = (const float*)d_in[3];
  const float* aw     = (const float*)d_in[4];
  float*       out    = (float*)d_out;

  // Structure constants from the reference: B=2, Hd=8, C=32, P=4, 4 levels.
  const long long B = 2, Hd = 8, C = 32, NL = 4, P = 4;
  const unsigned Q   = (unsigned)((long long)in_sizes[3] / (B * Hd * NL * P * 2));
  const unsigned L   = (unsigned)((long long)in_sizes[0] / (B * Hd * C));
  const unsigned nbq = (unsigned)(B * Q);

  // ~4 grid-stride iterations per block so the async double-buffer overlaps.
  unsigned grid = (nbq + 3u) / 4u;
  if (grid == 0) grid = 1;

  MSDeformAttn_gfx1250_kernel<<<grid, 256, 0, stream>>>(
      value, shapes, starts, loc, aw, out, Q, L, nbq);
}